// ConvolutionalLayer_26963804684983
// MI455X (gfx1250) — compile-verified
//
#include <hip/hip_runtime.h>

// SNN layer fused kernel for MI455X (gfx1250, wave32).
// conv2d(5x5x2->64, SAME) via V_WMMA_F32_16X16X4_F32 implicit GEMM,
// fused with potential accumulate, spike threshold, lateral inhibition
// (algebraically simplified, exact), and k_inh update.
//
// Round-2 changes:
//  - branch-free K padding: k=50/51 decode to xs row 5, which is zero-filled,
//    eliminating the saveexec/predicated-ds_load tail in the WMMA chain.
//  - 64-pixel tile per block: each wave computes two M-subtiles sharing one
//    B fragment (26 WMMAs/wave, half the weight-staging traffic).
//  - global_prefetch of the old_potentials tile to overlap HBM latency.

typedef float v2f __attribute__((ext_vector_type(2)));
typedef float v8f __attribute__((ext_vector_type(8)));

#define H       512
#define W       512
#define CIN     2
#define COUT    64
#define KTOT    50      // 5*5*2
#define KPAD    52      // pad K to multiple of 4 for 16x16x4 steps
#define TILE_W  64      // pixels per workgroup (one row segment)
#define XCOLS   (TILE_W + 4)
#define THRESH  10.0f

__global__ __launch_bounds__(256)
void snn_conv_spike_kernel(const float* __restrict__ x,     // [512,512,2]
                           const float* __restrict__ wgt,   // [5,5,2,64] HWIO
                           const float* __restrict__ oldp,  // [512,512,64]
                           const int*   __restrict__ kinh,  // [512,512]
                           float* __restrict__ Sout,        // [512,512,64]
                           float* __restrict__ Vout,        // [512,512,64]
                           int*   __restrict__ Kout)        // [512,512]
{
    __shared__ float Bs[KPAD][COUT];        // 52*64*4  = 13312 B, rows 50..51 zero
    __shared__ float xs[6][XCOLS][CIN];     // 6*68*2*4 =  3264 B, row 5 zero (K pad)
    __shared__ int   spikeflag[TILE_W];

    const int tid   = threadIdx.x;
    const int h     = blockIdx.y;
    const int wbase = blockIdx.x * TILE_W;

    // ---- prefetch this block's old_potentials tile (64 px * 64 ch * 4B = 16KB),
    //      one 64B line per thread, overlapped with LDS staging + WMMA ----
    {
        const float* opbase = oldp + ((size_t)h * W + wbase) * COUT;
        __builtin_prefetch(opbase + (size_t)tid * 16, 0, 1);
    }

    // ---- stage weights into LDS, zero-padded to K=52 ----
    // flat HWIO index ((ky*5+kx)*2+ic)*64 + oc == k*64 + oc
    for (int idx = tid; idx < KPAD * COUT; idx += 256) {
        int k = idx >> 6;
        Bs[k][idx & 63] = (k < KTOT) ? wgt[idx] : 0.0f;
    }
    // ---- stage x halo (rows h-2..h+2, cols wbase-2..wbase+65), zero-padded.
    //      Row 5 is an all-zero pad row: K=50/51 decode to ky=5 with no branch. ----
    for (int e = tid; e < 6 * XCOLS * CIN; e += 256) {
        int ic  = e & 1;
        int t   = e >> 1;
        int col = t % XCOLS;
        int row = t / XCOLS;
        int gh  = h + row - 2;
        int gw  = wbase + col - 2;
        float v = 0.0f;
        if (row < 5 && (unsigned)gh < (unsigned)H && (unsigned)gw < (unsigned)W)
            v = x[((size_t)gh * W + gw) * CIN + ic];
        xs[row][col][ic] = v;
    }
    if (tid < TILE_W) spikeflag[tid] = 0;
    __syncthreads();

    const int lane  = tid & 31;
    const int wave  = tid >> 5;
    const int nwave = wave & 3;    // 4 waves cover 64 channels (N tiles of 16)
    const int mwave = wave >> 2;   // 2 waves x 2 subtiles cover 64 pixels
    const int half  = lane >> 4;   // lane half (VGPR-layout K split)
    const int l16   = lane & 15;

    const int mcol0 = mwave * 16 + l16;        // pixel col (subtile 0) in xs
    const int mcol1 = (mwave + 2) * 16 + l16;  // pixel col (subtile 1) in xs

    // ---- implicit GEMM: 13 K-steps x 2 M-subtiles = 26 WMMAs per wave.
    //      B fragment shared by both subtiles. Branch-free K padding. ----
    v8f acc0 = {}, acc1 = {};
    #pragma unroll
    for (int kb = 0; kb < KPAD; kb += 4) {
        const int k0 = kb + 2 * half;   // layout: VGPR j holds K = kb + 2*half + j
        const int k1 = k0 + 1;

        // decode k -> (ky, kx, ic); k=50/51 -> ky=5 (zero row), no guard needed
        const int t0 = k0 >> 1, ic0 = k0 & 1;
        const int ky0 = t0 / 5, kx0 = t0 - ky0 * 5;
        const int t1 = k1 >> 1, ic1 = k1 & 1;
        const int ky1 = t1 / 5, kx1 = t1 - ky1 * 5;

        v2f b;                                   // channel n = l16
        b.x = Bs[k0][nwave * 16 + l16];
        b.y = Bs[k1][nwave * 16 + l16];

        v2f a0, a1;                              // pixel m = l16 per subtile
        a0.x = xs[ky0][mcol0 + kx0][ic0];
        a0.y = xs[ky1][mcol0 + kx1][ic1];
        a1.x = xs[ky0][mcol1 + kx0][ic0];
        a1.y = xs[ky1][mcol1 + kx1][ic1];

        acc0 = __builtin_amdgcn_wmma_f32_16x16x4_f32(
            false, a0, false, b, (short)0, acc0, false, false);
        acc1 = __builtin_amdgcn_wmma_f32_16x16x4_f32(
            false, a1, false, b, (short)0, acc1, false, false);
    }

    // ---- fused epilogue ----
    // C/D layout: VGPR r, lane -> pixel m = r + 8*half, channel n = l16.
    // Exact simplification of reference WTA (V is reset before the max, so a
    // spiking channel never "loses"):
    //   S_out = spiked && (k_inh != 0);  V = spiked ? 0 : pot;
    //   k_inh_new = (any-spike && k_inh) ? 0 : k_inh.
    const int ch = nwave * 16 + l16;
    #pragma unroll
    for (int t = 0; t < 2; ++t) {
        const int msub = mwave + 2 * t;
        #pragma unroll
        for (int r = 0; r < 8; ++r) {
            int   m   = r + 8 * half;
            int   pin = msub * 16 + m;               // pixel within block
            size_t p  = (size_t)h * W + wbase + pin; // global pixel
            float conv = t ? acc1[r] : acc0[r];
            float pot  = conv + oldp[p * COUT + ch];
            bool  s    = pot >= THRESH;
            bool  kin  = kinh[p] != 0;
            Vout[p * COUT + ch] = s ? 0.0f : pot;
            Sout[p * COUT + ch] = (s && kin) ? 1.0f : 0.0f;
            if (s) atomicOr(&spikeflag[pin], 1);
        }
    }
    __syncthreads();

    if (tid < TILE_W) {
        size_t p = (size_t)h * W + wbase + tid;
        int kv = kinh[p];
        Kout[p] = (spikeflag[tid] && kv) ? 0 : kv;
    }
}

extern "C" void kernel_launch(void* const* d_in, const int* in_sizes, int n_in,
                              void* d_out, int out_size, void* d_ws, size_t ws_size,
                              hipStream_t stream) {
    (void)in_sizes; (void)n_in; (void)out_size; (void)d_ws; (void)ws_size;

    const float* x    = (const float*)d_in[0];  // [1,512,512,2]
    const float* wgt  = (const float*)d_in[1];  // [5,5,2,64]
    const float* oldp = (const float*)d_in[2];  // [1,512,512,64]
    const int*   kinh = (const int*)d_in[3];    // [512,512]

    float* Sout = (float*)d_out;                          // [1,512,512,64]
    float* Vout = Sout + (size_t)H * W * COUT;            // [1,512,512,64]
    int*   Kout = (int*)(Vout + (size_t)H * W * COUT);    // [512,512] int32 bits

    dim3 grid(W / TILE_W, H, 1);   // 8 x 512 workgroups
    snn_conv_spike_kernel<<<grid, 256, 0, stream>>>(x, wgt, oldp, kinh, Sout, Vout, Kout);
}